// Embedding_14233521619455
// MI455X (gfx1250) — compile-verified
//
#include <hip/hip_runtime.h>
#include <stdint.h>

// Embedding gather: out[n, :] = W[:, tokens[n]] + bias,  W is [E=768, V=50257] row-major f32.
// Memory-bound (~200 MB total traffic). CDNA5 path: async gather into LDS with a
// 4-deep ASYNCcnt-pipelined double... quad buffer, coalesced b128 LDS readback,
// nontemporal b128 output stores (keep W resident in the 192MB L2).

#define VOCAB   50257
#define EMBED   768
#define NTOK    16          // tokens per block
#define NBUF    4           // pipeline depth / LDS buffers
#define THREADS 192         // EMBED/4 lanes -> 6 full wave32s, EXEC all ones

typedef float v4f __attribute__((ext_vector_type(4)));

__device__ __forceinline__ void wait_asynccnt_le(int n) {
  // n is a compile-time constant after full unroll; switch keeps asm immediates literal.
  switch (n) {
    case 0:  asm volatile("s_wait_asynccnt 0"  ::: "memory"); break;
    case 4:  asm volatile("s_wait_asynccnt 4"  ::: "memory"); break;
    case 8:  asm volatile("s_wait_asynccnt 8"  ::: "memory"); break;
    default: asm volatile("s_wait_asynccnt 12" ::: "memory"); break;
  }
}

__global__ __launch_bounds__(THREADS)
void embed_gather_async(const int* __restrict__ tokens,
                        const float* __restrict__ W,
                        const float* __restrict__ bias,
                        float* __restrict__ out) {
  __shared__ __align__(16) float buf[NBUF][EMBED];   // 12 KB of the 320 KB WGP LDS

  const int tid     = threadIdx.x;        // 0..191
  const int tokBase = blockIdx.x * NTOK;
  const int e0      = tid * 4;            // this lane owns embedding rows e0..e0+3

  // Byte offsets of this lane's 4 rows within W (max ~154e6 < 2^31, fits u32).
  uint32_t row[4];
#pragma unroll
  for (int k = 0; k < 4; ++k)
    row[k] = (uint32_t)(e0 + k) * (uint32_t)VOCAB * 4u;

  // Bias held in registers, reused for all NTOK tokens.
  const v4f b4 = *(const v4f*)(bias + e0);

  // LDS byte address of this lane's slice in buffer 0 (generic ptr low 32 bits
  // are the workgroup-relative LDS address on gfx1250).
  const uint32_t lds0 = (uint32_t)(uintptr_t)&buf[0][0] + (uint32_t)tid * 16u;

  const int* tptr = tokens + tokBase;     // uniform -> scalar loads

  // Issue one pipeline stage: async-gather 4 scattered W words for token s
  // into this lane's 16B slice of buffer (s % NBUF). 4 ASYNCcnt increments/wave.
  auto issue = [&](int s) {
    const uint32_t tb = (uint32_t)tptr[s] * 4u;                       // token column byte offset
    const uint32_t l  = lds0 + (uint32_t)(s & (NBUF - 1)) * (EMBED * 4u);
#pragma unroll
    for (int k = 0; k < 4; ++k) {
      asm volatile("global_load_async_to_lds_b32 %0, %1, %2"
                   :: "v"(l + 4u * k),        // per-lane LDS dest address
                      "v"(row[k] + tb),       // per-lane 32-bit voffset
                      "s"(W)                  // 64-bit SGPR base
                   : "memory");
    }
  };

  // Prologue: fill the pipeline (stages 0..NBUF-2).
#pragma unroll
  for (int s = 0; s < NBUF - 1; ++s) issue(s);

  float* op = out + (size_t)tokBase * EMBED + e0;

#pragma unroll
  for (int i = 0; i < NTOK; ++i) {
    // Keep the async queue NBUF-1 stages ahead. Buffer (i+NBUF-1)%NBUF was last
    // read by stage i-1, whose loaded values were already consumed by its store.
    if (i + NBUF - 1 < NTOK) issue(i + NBUF - 1);

    // Wait until stage i's 4 async loads have landed in LDS (leave the
    // younger stages in flight).
    const int r = NTOK - 1 - i;
    wait_asynccnt_le(4 * (r < NBUF - 1 ? r : NBUF - 1));

    // Coalesced 16B LDS readback of our own wave's gathered data (no barrier
    // needed: lane slices are disjoint and written by this wave's async ops).
    v4f v = *(const v4f*)&buf[i & (NBUF - 1)][e0];
    v += b4;

    // Streaming output: nontemporal so 48 MB of writes don't evict W from L2.
    __builtin_nontemporal_store(v, (v4f*)(op + (size_t)i * EMBED));
  }
}

extern "C" void kernel_launch(void* const* d_in, const int* in_sizes, int n_in,
                              void* d_out, int out_size, void* d_ws, size_t ws_size,
                              hipStream_t stream) {
  const int*   tokens = (const int*)d_in[0];   // [B*T] int32
  const float* W      = (const float*)d_in[1]; // [E, V] f32
  const float* bias   = (const float*)d_in[2]; // [E] f32
  float*       out    = (float*)d_out;         // [B*T, E] f32

  const int ntok_total = in_sizes[0];          // 16384, divisible by NTOK
  const int blocks     = ntok_total / NTOK;    // 1024 blocks x 192 threads

  embed_gather_async<<<blocks, THREADS, 0, stream>>>(tokens, W, bias, out);
}